// CausalFMMAttention_15891378995446
// MI455X (gfx1250) — compile-verified
//
#include <hip/hip_runtime.h>
#include <math.h>
#include <stdint.h>

// MI455X / gfx1250, wave32. Chunked causal linear attention on
// V_WMMA_F32_16X16X4_F32 (exact f32 precision) + double-buffered
// GLOBAL_LOAD_ASYNC_TO_LDS_B128 prefetch of the next chunk (ASYNCcnt),
// which overlaps HBM latency with the serial per-chunk WMMA work.

typedef __attribute__((ext_vector_type(2))) float v2f;
typedef __attribute__((ext_vector_type(8))) float v8f;

#define N_ 2
#define L_ 2048
#define H_ 8
#define E_ 32
#define RS (H_*E_)          // float stride between consecutive sequence rows
#define CHUNK 16
#define NCHUNK (L_/CHUNK)
#define BAND 10
#define EPS_ 1e-6f

static __device__ __forceinline__ v8f wmma_f32(v2f a, v2f b, v8f c) {
  // D(16x16,f32) = A(16x4,f32) * B(4x16,f32) + C
  return __builtin_amdgcn_wmma_f32_16x16x4_f32(
      /*neg_a=*/false, a, /*neg_b=*/false, b,
      /*c_mod=*/(short)0, c, /*reuse_a=*/false, /*reuse_b=*/false);
}

static __device__ __forceinline__ float elu1(float x) {
  // elu(x)+1
  return x > 0.f ? x + 1.f : __expf(x);
}

static __device__ __forceinline__ v8f zero8() {
  v8f z = {0.f,0.f,0.f,0.f,0.f,0.f,0.f,0.f};
  return z;
}

// ---------------------------------------------------------------------------
// Kernel 1: banded softmax attention, out = W1 * softmax(band(QK^T/sqrt(E)))V
// Band width 10 => mostly-masked tiles; scalar VALU path. Writes every output
// element (initializes d_out after the harness poisons it).
// ---------------------------------------------------------------------------
__global__ __launch_bounds__(256)
void band_attn_kernel(const float* __restrict__ Q, const float* __restrict__ K,
                      const float* __restrict__ V, const float* __restrict__ W1,
                      float* __restrict__ out) {
  int idx = blockIdx.x * blockDim.x + threadIdx.x;
  if (idx >= N_ * L_ * H_) return;
  int h = idx % H_;
  int l = (idx / H_) % L_;
  int n = idx / (H_ * L_);
  const float temp = 0.17677669529663687f;  // 1/sqrt(32)
  const float* q = Q + ((size_t)(n * L_ + l) * H_ + h) * E_;
  int s0 = l - (BAND - 1); if (s0 < 0) s0 = 0;
  int cnt = l - s0 + 1;

  // pull the band window toward the caches early (global_prefetch_b8)
  __builtin_prefetch(K + ((size_t)(n * L_ + s0) * H_ + h) * E_, 0, 1);
  __builtin_prefetch(V + ((size_t)(n * L_ + s0) * H_ + h) * E_, 0, 1);

  float sc[BAND];
  float mx = -3.402823466e38f;
  for (int i = 0; i < cnt; ++i) {
    const float* k = K + ((size_t)(n * L_ + s0 + i) * H_ + h) * E_;
    float d = 0.f;
    #pragma unroll
    for (int e = 0; e < E_; ++e) d += q[e] * k[e];
    d *= temp;
    sc[i] = d;
    mx = fmaxf(mx, d);
  }
  float acc[E_];
  #pragma unroll
  for (int e = 0; e < E_; ++e) acc[e] = 0.f;
  float sum = 0.f;
  for (int i = 0; i < cnt; ++i) {
    float p = __expf(sc[i] - mx);
    sum += p;
    const float* v = V + ((size_t)(n * L_ + s0 + i) * H_ + h) * E_;
    #pragma unroll
    for (int e = 0; e < E_; ++e) acc[e] += p * v[e];
  }
  float inv = 1.f / sum;
  float* o = out + ((size_t)(n * L_ + l) * H_ + h) * E_;
  #pragma unroll
  for (int e = 0; e < E_; ++e) o[e] = W1[h * E_ + e] * acc[e] * inv;
}

// ---------------------------------------------------------------------------
// Async stage of one raw 16x32 chunk of Q, K, V into LDS (double buffer).
// Each lane owns one 16-float half-row => 4 x b128 per matrix, 12 async ops.
// LDS operand = low 32 bits of the generic shared-pointer (flat LDS aperture
// keeps the byte offset in addr[31:0], ISA 10.2).
// ---------------------------------------------------------------------------
static __device__ __forceinline__ void stage_chunk_async(
    const float* __restrict__ Qg, const float* __restrict__ Kg,
    const float* __restrict__ Vg, int l0,
    float* bq, float* bk, float* bv, int srow, int scol) {
  const size_t go = (size_t)(l0 + srow) * RS + scol;
  unsigned lq = (unsigned)(uintptr_t)(bq + (srow * E_ + scol));
  unsigned lk = (unsigned)(uintptr_t)(bk + (srow * E_ + scol));
  unsigned lv = (unsigned)(uintptr_t)(bv + (srow * E_ + scol));
  unsigned long long aq = (unsigned long long)(uintptr_t)(Qg + go);
  unsigned long long ak = (unsigned long long)(uintptr_t)(Kg + go);
  unsigned long long av = (unsigned long long)(uintptr_t)(Vg + go);
  #pragma unroll
  for (int j = 0; j < 4; ++j) {
    unsigned b = 16u * (unsigned)j;
    asm volatile("global_load_async_to_lds_b128 %0, %1, off"
                 :: "v"(lq + b), "v"(aq + (unsigned long long)b) : "memory");
    asm volatile("global_load_async_to_lds_b128 %0, %1, off"
                 :: "v"(lk + b), "v"(ak + (unsigned long long)b) : "memory");
    asm volatile("global_load_async_to_lds_b128 %0, %1, off"
                 :: "v"(lv + b), "v"(av + (unsigned long long)b) : "memory");
  }
}

// ---------------------------------------------------------------------------
// Kernel 2: two causal linear-attention branches (phi = elu+1 and (elu+1)^2),
// chunked-scan formulation, all matmuls on V_WMMA_F32_16X16X4_F32.
// One wave (32 threads) per (n,h); branches processed sequentially; output is
// accumulated via exclusive-ownership read-modify-write (deterministic).
// key_lengths cancels exactly under the per-row K normalization.
// ---------------------------------------------------------------------------
__global__ __launch_bounds__(32)
void linear_attn_kernel(const float* __restrict__ Q, const float* __restrict__ K,
                        const float* __restrict__ V, const float* __restrict__ W2,
                        const float* __restrict__ W3, float* __restrict__ out) {
  __shared__ __align__(16) float rQ[2][CHUNK * E_]; // raw Q chunk, double buffer
  __shared__ __align__(16) float rK[2][CHUNK * E_]; // raw K chunk
  __shared__ __align__(16) float rV[2][CHUNK * E_]; // raw V chunk
  __shared__ float sQ[CHUNK * E_];   // normalized phi(Q) chunk, 16x32
  __shared__ float sK[CHUNK * E_];   // normalized phi(K) chunk, 16x32
  __shared__ float sS[E_ * E_];      // running state S_prev staging, 32x32
  __shared__ float sA[CHUNK * CHUNK];// masked QK^T, 16x16
  __shared__ float sKcum[E_];        // running key cumsum (32)
  __shared__ float sInv[2 * CHUNK];  // row-sum reciprocals (Q rows, K rows)
  __shared__ float sZ[CHUNK];        // per-row 1/denominator

  const int wg = blockIdx.x;         // 0..N*H-1
  const int h = wg % H_;
  const int n = wg / H_;
  const int t = threadIdx.x;         // 0..31
  const int tlo = t & 15;
  const int thi = t >> 4;            // half-wave select

  const float* Qg = Q + ((size_t)n * L_ * H_ + h) * E_;
  const float* Kg = K + ((size_t)n * L_ * H_ + h) * E_;
  const float* Vg = V + ((size_t)n * L_ * H_ + h) * E_;
  float* Og = out + ((size_t)n * L_ * H_ + h) * E_;

  const int srow = t >> 1;           // staging: each lane owns half a row
  const int scol = (t & 1) * 16;

  #pragma unroll 1
  for (int branch = 0; branch < 2; ++branch) {
    const float* Wb = (branch == 0) ? W2 : W3;
    const float w0 = Wb[h * E_ + tlo];
    const float w1 = Wb[h * E_ + 16 + tlo];

    // state accumulators: 2x2 tiles of 16x16 (C/D layout)
    v8f S00 = zero8(), S01 = zero8(), S10 = zero8(), S11 = zero8();
    sKcum[t] = 0.f;

    // prologue: stage chunk 0 into buffer 0
    stage_chunk_async(Qg, Kg, Vg, 0, rQ[0], rK[0], rV[0], srow, scol);

    #pragma unroll 1
    for (int c = 0; c < NCHUNK; ++c) {
      const int l0 = c * CHUNK;
      const float* bq = rQ[c & 1];
      const float* bk = rK[c & 1];
      const float* bv = rV[c & 1];

      // prefetch next chunk into the other buffer, then wait for this one.
      // Async loads complete in order: <=12 outstanding => stage(c) landed.
      if (c + 1 < NCHUNK) {
        stage_chunk_async(Qg, Kg, Vg, l0 + CHUNK,
                          rQ[(c + 1) & 1], rK[(c + 1) & 1], rV[(c + 1) & 1],
                          srow, scol);
        asm volatile("s_wait_asynccnt 0xc" ::: "memory");
      } else {
        asm volatile("s_wait_asynccnt 0x0" ::: "memory");
      }

      // ---- transform raw -> phi(Q), phi(K) in LDS ----
      #pragma unroll
      for (int i = 0; i < 16; ++i) {
        float fq = elu1(bq[srow * E_ + scol + i]); if (branch) fq *= fq;
        float fk = elu1(bk[srow * E_ + scol + i]); if (branch) fk *= fk;
        sQ[srow * E_ + scol + i] = fq;
        sK[srow * E_ + scol + i] = fk;
      }
      __syncthreads();
      {  // per-row normalization: lanes 0-15 do Q rows, 16-31 do K rows
        const float* base = (thi ? sK : sQ) + tlo * E_;
        float s = 0.f;
        #pragma unroll
        for (int e = 0; e < E_; ++e) s += base[e];
        sInv[t] = 1.f / s;
      }
      __syncthreads();
      {
        float iq = sInv[srow];
        float ik = sInv[16 + srow];
        #pragma unroll
        for (int i = 0; i < 16; ++i) {
          sQ[srow * E_ + scol + i] *= iq;
          sK[srow * E_ + scol + i] *= ik;
        }
      }
      __syncthreads();

      // ---- build WMMA fragments (ISA 7.12.2 layouts) ----
      // A 16x4: lane m=tlo, VGPR0/1 hold K = 4kk+2*thi {+0,+1}
      // B 4x16: lane n=tlo, VGPR0/1 hold rows K = 4kk+2*thi {+0,+1}
      v2f qf[8], kf[8];
      #pragma unroll
      for (int kk = 0; kk < 8; ++kk) {
        int col = 4 * kk + 2 * thi;
        qf[kk][0] = sQ[tlo * E_ + col]; qf[kk][1] = sQ[tlo * E_ + col + 1];
        kf[kk][0] = sK[tlo * E_ + col]; kf[kk][1] = sK[tlo * E_ + col + 1];
      }
      v2f vf[4][2];  // V as B-operand: rows = chunk positions, cols = features
      #pragma unroll
      for (int kk = 0; kk < 4; ++kk)
        #pragma unroll
        for (int nb = 0; nb < 2; ++nb) {
          int r0 = 4 * kk + 2 * thi;
          int cv = nb * 16 + tlo;
          vf[kk][nb][0] = bv[r0 * E_ + cv];
          vf[kk][nb][1] = bv[(r0 + 1) * E_ + cv];
        }
      v2f ktf[2][4]; // K^T as A-operand: m = feature (within kb), K = chunk row
      #pragma unroll
      for (int kb = 0; kb < 2; ++kb)
        #pragma unroll
        for (int kk = 0; kk < 4; ++kk) {
          int fcol = kb * 16 + tlo;
          int r0 = 4 * kk + 2 * thi;
          ktf[kb][kk][0] = sK[r0 * E_ + fcol];
          ktf[kb][kk][1] = sK[(r0 + 1) * E_ + fcol];
        }

      // ---- QK^T : 16x16, K-dim 32 => 8 WMMAs ----
      v8f acc = zero8();
      #pragma unroll
      for (int kk = 0; kk < 8; ++kk) acc = wmma_f32(qf[kk], kf[kk], acc);

      // causal mask, stage masked tile; stage S_prev (C layout -> row major)
      #pragma unroll
      for (int r = 0; r < 8; ++r) {
        int m = r + 8 * thi;
        sA[m * 16 + tlo] = (tlo <= m) ? acc[r] : 0.f;
      }
      #pragma unroll
      for (int r = 0; r < 8; ++r) {
        int m = r + 8 * thi;
        sS[m * E_ + tlo]             = S00[r];
        sS[m * E_ + 16 + tlo]        = S01[r];
        sS[(16 + m) * E_ + tlo]      = S10[r];
        sS[(16 + m) * E_ + 16 + tlo] = S11[r];
      }
      __syncthreads();

      // z[m] = 1/(q[m].Kcum_prev + rowsum(tril QK^T)[m] + eps)
      if (t < 16) {
        float sint = 0.f;
        #pragma unroll
        for (int nn = 0; nn < 16; ++nn) sint += sA[t * 16 + nn];
        float qd = 0.f;
        #pragma unroll
        for (int e = 0; e < E_; ++e) qd += sQ[t * E_ + e] * sKcum[e];
        sZ[t] = 1.f / (qd + sint + EPS_);
      }

      // ---- out = Q * S_prev  (K-dim 32) + tril(QK^T) * V (K-dim 16) ----
      v8f o0 = zero8(), o1 = zero8();
      #pragma unroll
      for (int k4 = 0; k4 < 8; ++k4) {
        int r0 = 4 * k4 + 2 * thi;
        v2f b0, b1;
        b0[0] = sS[r0 * E_ + tlo];      b0[1] = sS[(r0 + 1) * E_ + tlo];
        b1[0] = sS[r0 * E_ + 16 + tlo]; b1[1] = sS[(r0 + 1) * E_ + 16 + tlo];
        o0 = wmma_f32(qf[k4], b0, o0);
        o1 = wmma_f32(qf[k4], b1, o1);
      }
      #pragma unroll
      for (int kk = 0; kk < 4; ++kk) {
        int colA = 4 * kk + 2 * thi;
        v2f af;
        af[0] = sA[tlo * 16 + colA]; af[1] = sA[tlo * 16 + colA + 1];
        o0 = wmma_f32(af, vf[kk][0], o0);
        o1 = wmma_f32(af, vf[kk][1], o1);
      }

      // ---- S += K^T V  (4 tiles x 4 WMMAs) ----
      #pragma unroll
      for (int kk = 0; kk < 4; ++kk) {
        S00 = wmma_f32(ktf[0][kk], vf[kk][0], S00);
        S01 = wmma_f32(ktf[0][kk], vf[kk][1], S01);
        S10 = wmma_f32(ktf[1][kk], vf[kk][0], S10);
        S11 = wmma_f32(ktf[1][kk], vf[kk][1], S11);
      }
      __syncthreads();

      // Kcum += column sums of the normalized K chunk
      {
        float s = 0.f;
        #pragma unroll
        for (int m = 0; m < 16; ++m) s += sK[m * E_ + t];
        sKcum[t] += s;
      }
      // scale by z and per-(h,e) weight; accumulate into output (exclusive)
      #pragma unroll
      for (int r = 0; r < 8; ++r) {
        int m = r + 8 * thi;
        float z = sZ[m];
        float* o = Og + (size_t)(l0 + m) * RS + tlo;
        o[0]  += w0 * z * o0[r];
        o[16] += w1 * z * o1[r];
      }
      __syncthreads();
    }
  }
}

extern "C" void kernel_launch(void* const* d_in, const int* in_sizes, int n_in,
                              void* d_out, int out_size, void* d_ws, size_t ws_size,
                              hipStream_t stream) {
  (void)in_sizes; (void)n_in; (void)out_size; (void)d_ws; (void)ws_size;
  const float* Q  = (const float*)d_in[0];
  const float* K  = (const float*)d_in[1];
  const float* V  = (const float*)d_in[2];
  // d_in[3] = key_lengths: cancels under _norm_last(K) for nonzero lengths.
  const float* W1 = (const float*)d_in[4];
  const float* W2 = (const float*)d_in[5];
  const float* W3 = (const float*)d_in[6];
  float* out = (float*)d_out;

  // 1) banded softmax branch writes W1*SV (full initialization of d_out)
  band_attn_kernel<<<dim3((N_ * L_ * H_ + 255) / 256), dim3(256), 0, stream>>>(
      Q, K, V, W1, out);
  // 2) both linear branches accumulate W2*LV1 + W3*LV2 (same stream => ordered)
  linear_attn_kernel<<<dim3(N_ * H_), dim3(32), 0, stream>>>(
      Q, K, V, W2, W3, out);
}